// GATLSTM_46213848105969
// MI455X (gfx1250) — compile-verified
//
#include <hip/hip_runtime.h>
#include <math.h>

#define NN   10000
#define EE   160000
#define FF   32
#define DEE  8
#define HID  64
#define TT   3
#define OUTD 12
#define EPN  (EE + NN)          /* 170000 */
#define NEG_SLOPE 0.2f
#define LN_EPS    1e-5f

typedef float v2f __attribute__((ext_vector_type(2)));
typedef float v8f __attribute__((ext_vector_type(8)));

// ---------------------------------------------------------------------------
// float atomic max via sign-partitioned integer atomics (init bytes = 0xFF)
// ---------------------------------------------------------------------------
__device__ __forceinline__ void atomicMaxF(float* addr, float v) {
  if (v >= 0.0f)
    atomicMax((int*)addr, __float_as_int(v));
  else
    atomicMin((unsigned int*)addr, __float_as_uint(v));
}

// ---------------------------------------------------------------------------
// Unguarded WMMA fp32 GEMM for NCOLS == 64 (all tiles in-bounds by grid
// construction; nrows multiple of 16):
//   C[z][nrows x 64] = A[nrows x K] @ W[z][K x 64] + b[z]
// One wave computes a 16x32 tile: one A-fragment load feeds TWO
// V_WMMA_F32_16X16X4_F32 per k-step (2 independent accumulator chains).
// Fragment layout per CDNA5 ISA 7.12.2:
//   A: lanes 0-15 hold M=0..15 K={k0,k0+1}; lanes 16-31 hold K={k0+2,k0+3}
//   B: lanes 0-15 hold N=0..15 K={k0,k0+1}; lanes 16-31 hold K={k0+2,k0+3}
//   C/D: VGPR i -> M=i (lanes 0-15, N=lane), M=8+i (lanes 16-31, N=lane-16)
// ---------------------------------------------------------------------------
template <int K>
__global__ __launch_bounds__(32)
void wmma_gemm64_kernel(const float* __restrict__ A, const float* __restrict__ W,
                        const float* __restrict__ bias, float* __restrict__ C,
                        long wstride, long bstride, long ostride) {
  const int lane = threadIdx.x & 31;
  const int half = lane >> 4;       // 0: K pair {0,1}, 1: K pair {2,3}
  const int lm   = lane & 15;
  const int tm   = blockIdx.x * 16;
  const int tn   = blockIdx.y * 32; // two 16-col tiles per wave
  const int z    = blockIdx.z;
  const float* Wz = W    + (long)z * wstride + tn + lm;
  const float* bz = bias + (long)z * bstride + tn + lm;
  float*       Cz = C    + (long)z * ostride + tn + lm;

  // per-lane A base: row tm+lm, starting at K offset half*2 (8B aligned)
  const float* ar = A + (long)(tm + lm) * K + half * 2;

  v8f acc0 = {0.f,0.f,0.f,0.f,0.f,0.f,0.f,0.f};
  v8f acc1 = {0.f,0.f,0.f,0.f,0.f,0.f,0.f,0.f};
#pragma unroll
  for (int k0 = 0; k0 < K; k0 += 4) {
    const v2f a = *(const v2f*)(ar + k0);      // one b64 load, reused twice
    const int ka = k0 + half * 2;
    const float* w0 = Wz + (long)ka * HID;     // W[ka][tn+lm + {0,16}]
    v2f b0, b1;
    b0[0] = w0[0];   b0[1] = w0[HID];
    b1[0] = w0[16];  b1[1] = w0[HID + 16];
    acc0 = __builtin_amdgcn_wmma_f32_16x16x4_f32(
        false, a, false, b0, (short)0, acc0, false, false);
    acc1 = __builtin_amdgcn_wmma_f32_16x16x4_f32(
        false, a, false, b1, (short)0, acc1, false, false);
  }

  const float bv0 = bz[0];
  const float bv1 = bz[16];
#pragma unroll
  for (int i = 0; i < 8; ++i) {
    float* crow = Cz + (long)(tm + half * 8 + i) * HID;
    crow[0]  = acc0[i] + bv0;
    crow[16] = acc1[i] + bv1;
  }
}

// ---------------------------------------------------------------------------
// Guarded WMMA GEMM for the final 64 -> 12 projection (NCOLS not a multiple
// of 16): single 16x16 tile per wave, B/bias/stores masked to col < NCOLS.
// ---------------------------------------------------------------------------
template <int K, int NCOLS>
__global__ __launch_bounds__(32)
void wmma_gemm_small_kernel(const float* __restrict__ A, const float* __restrict__ W,
                            const float* __restrict__ bias, float* __restrict__ C) {
  const int lane = threadIdx.x & 31;
  const int half = lane >> 4;
  const int lm   = lane & 15;
  const int tm   = blockIdx.x * 16;
  const int  col   = lm;             // single col-tile at origin (NCOLS <= 16)
  const bool colok = (col < NCOLS);

  const float* ar = A + (long)(tm + lm) * K + half * 2;
  v8f acc = {0.f,0.f,0.f,0.f,0.f,0.f,0.f,0.f};
#pragma unroll
  for (int k0 = 0; k0 < K; k0 += 4) {
    const v2f a = *(const v2f*)(ar + k0);
    const int ka = k0 + half * 2;
    v2f b;
    if (colok) {
      b[0] = W[(long)ka * NCOLS + col];
      b[1] = W[(long)(ka + 1) * NCOLS + col];
    } else {
      b[0] = 0.f; b[1] = 0.f;
    }
    acc = __builtin_amdgcn_wmma_f32_16x16x4_f32(
        false, a, false, b, (short)0, acc, false, false);
  }
  if (colok) {
    const float bv = bias[col];
#pragma unroll
    for (int i = 0; i < 8; ++i) {
      const int row = tm + half * 8 + i;
      C[(long)row * NCOLS + col] = acc[i] + bv;
    }
  }
}

// ---------------------------------------------------------------------------
// Self-loop edge_attr: degree + sum of incoming edge_attr per dst
// ---------------------------------------------------------------------------
__global__ __launch_bounds__(256)
void deg_accum_kernel(const int* __restrict__ ei, const float* __restrict__ ea,
                      float* __restrict__ deg, float* __restrict__ sumea) {
  const int e = blockIdx.x * 256 + threadIdx.x;
  if (e >= EE) return;
  const int d = ei[EE + e];
  atomicAdd(&deg[d], 1.0f);
#pragma unroll
  for (int k = 0; k < DEE; ++k)
    atomicAdd(&sumea[(long)d * DEE + k], ea[(long)e * DEE + k]);
}

__global__ __launch_bounds__(256)
void mean_div_kernel(float* __restrict__ mean_ea, const float* __restrict__ deg) {
  const int i = blockIdx.x * 256 + threadIdx.x;
  if (i >= NN * DEE) return;
  mean_ea[i] /= fmaxf(deg[i / DEE], 1.0f);
}

// ---------------------------------------------------------------------------
// Edge pass 1 (wave-per-edge): s = leaky(xl[src]+xr[dst]+ea@We);
// logit_h = s . att_h via __shfl_xor wave reduction; atomicMax per-dst max.
// Lane l owns channels l and l+32 (head 0 / head 1). grid: (edges/8, gate, br)
// ---------------------------------------------------------------------------
__global__ __launch_bounds__(256)
void edge_logits_kernel(const int* __restrict__ ei,
                        const float* __restrict__ edge_attr,
                        const float* __restrict__ mean_ea,
                        const float* __restrict__ xlx, const float* __restrict__ xrx,
                        const float* __restrict__ hlb, const float* __restrict__ hrb,
                        const float* __restrict__ We_x, const float* __restrict__ We_h,
                        const float* __restrict__ att_x, const float* __restrict__ att_h,
                        float* __restrict__ logits, float* __restrict__ mbuf) {
  __shared__ float sWe[DEE * HID];
  __shared__ float sAtt[HID];
  const int g = blockIdx.y, br = blockIdx.z;
  const float* We  = (br ? We_h  : We_x)  + (long)g * DEE * HID;
  const float* att = (br ? att_h : att_x) + (long)g * HID;
  for (int i = threadIdx.x; i < DEE * HID; i += 256) sWe[i] = We[i];
  if (threadIdx.x < HID) sAtt[threadIdx.x] = att[threadIdx.x];
  __syncthreads();

  const int  lane = threadIdx.x & 31;
  const long e    = (long)blockIdx.x * 8 + (threadIdx.x >> 5);
  if (e >= EPN) return;
  int s, d; const float* eav;
  if (e < EE) { s = ei[e]; d = ei[EE + e]; eav = edge_attr + e * DEE; }
  else        { s = d = (int)(e - EE);     eav = mean_ea + (e - EE) * DEE; }

  const float* xl = (br ? hlb : xlx) + ((long)g * NN + s) * HID;
  const float* xr = (br ? hrb : xrx) + ((long)g * NN + d) * HID;

  float ea[DEE];
#pragma unroll
  for (int k = 0; k < DEE; ++k) ea[k] = eav[k];   // 32B broadcast read

  const int c0 = lane, c1 = lane + 32;
  float ev0 = 0.f, ev1 = 0.f;
#pragma unroll
  for (int k = 0; k < DEE; ++k) {
    ev0 = fmaf(ea[k], sWe[k * HID + c0], ev0);
    ev1 = fmaf(ea[k], sWe[k * HID + c1], ev1);
  }
  float s0 = xl[c0] + xr[c0] + ev0;               // coalesced 128B row reads
  float s1 = xl[c1] + xr[c1] + ev1;
  s0 = (s0 >= 0.f) ? s0 : NEG_SLOPE * s0;
  s1 = (s1 >= 0.f) ? s1 : NEG_SLOPE * s1;
  float l0 = sAtt[c0] * s0;                       // head 0 partial (c0 < 32)
  float l1 = sAtt[c1] * s1;                       // head 1 partial (c1 >= 32)
#pragma unroll
  for (int off = 16; off >= 1; off >>= 1) {
    l0 += __shfl_xor(l0, off, 32);
    l1 += __shfl_xor(l1, off, 32);
  }
  if (lane == 0) {
    const int  bg    = br * 4 + g;
    const long lbase = ((long)bg * EPN + e) * 2;
    logits[lbase]     = l0;
    logits[lbase + 1] = l1;
    atomicMaxF(&mbuf[((long)bg * NN + d) * 2],     l0);
    atomicMaxF(&mbuf[((long)bg * NN + d) * 2 + 1], l1);
  }
}

// ---------------------------------------------------------------------------
// Edge pass 2 (wave-per-edge): w = exp(logit - m[dst]); z[dst]+=w;
// msg[dst][c] += w_h * xl[src][c]  -- 2 coalesced atomics per lane.
// ---------------------------------------------------------------------------
__global__ __launch_bounds__(256)
void edge_accum_kernel(const int* __restrict__ ei,
                       const float* __restrict__ logits,
                       const float* __restrict__ mbuf,
                       const float* __restrict__ xlx, const float* __restrict__ hlb,
                       float* __restrict__ zbuf, float* __restrict__ msg) {
  const int g = blockIdx.y, br = blockIdx.z;
  const int  lane = threadIdx.x & 31;
  const long e    = (long)blockIdx.x * 8 + (threadIdx.x >> 5);
  if (e >= EPN) return;
  int s, d;
  if (e < EE) { s = ei[e]; d = ei[EE + e]; } else { s = d = (int)(e - EE); }
  const int  bg    = br * 4 + g;
  const long lbase = ((long)bg * EPN + e) * 2;
  const long nb    = ((long)bg * NN + d) * 2;
  const float w0 = expf(logits[lbase]     - mbuf[nb]);      // broadcast loads
  const float w1 = expf(logits[lbase + 1] - mbuf[nb + 1]);
  if (lane == 0) {
    atomicAdd(&zbuf[nb],     w0);
    atomicAdd(&zbuf[nb + 1], w1);
  }
  const float* xl = (br ? hlb : xlx) + ((long)g * NN + s) * HID;
  float* mrow = msg + ((long)bg * NN + d) * HID;
  atomicAdd(&mrow[lane],      w0 * xl[lane]);
  atomicAdd(&mrow[lane + 32], w1 * xl[lane + 32]);
}

// ---------------------------------------------------------------------------
// Normalize messages by softmax denominator, add per-gate bias
// ---------------------------------------------------------------------------
__global__ __launch_bounds__(256)
void msg_finalize_kernel(float* __restrict__ msg, const float* __restrict__ zbuf,
                         const float* __restrict__ bias_x,
                         const float* __restrict__ bias_h) {
  const long i = (long)blockIdx.x * 256 + threadIdx.x;
  if (i >= 8L * NN * HID) return;
  const int  c  = (int)(i % HID);
  const long nn = i / HID;           // bg*NN + n
  const int  bg = (int)(nn / NN);
  const int  g  = bg & 3, br = bg >> 2;
  const float* bias = (br ? bias_h : bias_x) + g * HID;
  const float z = zbuf[nn * 2 + (c >> 5)];
  msg[i] = msg[i] / (z + 1e-16f) + bias[c];
}

// ---------------------------------------------------------------------------
// LSTM gate combine + LayerNorm. Wave-per-node, 2 channels/lane, shuffle LN.
// ---------------------------------------------------------------------------
__global__ __launch_bounds__(256)
void lstm_kernel(const float* __restrict__ msg, float* __restrict__ cstate,
                 float* __restrict__ hstate, const float* __restrict__ ln_w,
                 const float* __restrict__ ln_b) {
  const int lane = threadIdx.x & 31;
  const int n    = blockIdx.x * 8 + (threadIdx.x >> 5);
  if (n >= NN) return;
  float pre[2];
  float sum = 0.f;
#pragma unroll
  for (int j = 0; j < 2; ++j) {
    const int  c = lane + j * 32;
    const long o = (long)n * HID + c;
    float iv = msg[0L * NN * HID + o] + msg[4L * NN * HID + o];
    float fv = msg[1L * NN * HID + o] + msg[5L * NN * HID + o];
    float ov = msg[2L * NN * HID + o] + msg[6L * NN * HID + o];
    float gv = msg[3L * NN * HID + o] + msg[7L * NN * HID + o];
    iv = 1.f / (1.f + expf(-iv));
    fv = 1.f / (1.f + expf(-fv));
    ov = 1.f / (1.f + expf(-ov));
    gv = tanhf(gv);
    const float cn = fv * cstate[o] + iv * gv;
    cstate[o] = cn;
    const float p = ov * tanhf(cn);
    pre[j] = p;
    sum += p;
  }
#pragma unroll
  for (int off = 16; off >= 1; off >>= 1) sum += __shfl_xor(sum, off, 32);
  const float mu = sum * (1.f / HID);
  float var = 0.f;
#pragma unroll
  for (int j = 0; j < 2; ++j) { const float dv = pre[j] - mu; var += dv * dv; }
#pragma unroll
  for (int off = 16; off >= 1; off >>= 1) var += __shfl_xor(var, off, 32);
  var *= (1.f / HID);
  const float inv = rsqrtf(var + LN_EPS);
#pragma unroll
  for (int j = 0; j < 2; ++j) {
    const int c = lane + j * 32;
    hstate[(long)n * HID + c] = (pre[j] - mu) * inv * ln_w[c] + ln_b[c];
  }
}

// ---------------------------------------------------------------------------
// Exact-erf GELU (matches torch GELU / jax approximate=False)
// ---------------------------------------------------------------------------
__global__ __launch_bounds__(256)
void gelu_kernel(float* __restrict__ t) {
  const long i = (long)blockIdx.x * 256 + threadIdx.x;
  if (i >= (long)NN * HID) return;
  const float x = t[i];
  t[i] = 0.5f * x * (1.f + erff(x * 0.70710678118654752440f));
}

// ---------------------------------------------------------------------------
extern "C" void kernel_launch(void* const* d_in, const int* in_sizes, int n_in,
                              void* d_out, int out_size, void* d_ws, size_t ws_size,
                              hipStream_t stream) {
  (void)in_sizes; (void)n_in; (void)out_size; (void)ws_size;
  const float* x_seq     = (const float*)d_in[0];
  const float* edge_attr = (const float*)d_in[1];
  const int*   ei        = (const int*)  d_in[2];
  const float* Wl_x   = (const float*)d_in[3];
  const float* bl_x   = (const float*)d_in[4];
  const float* Wr_x   = (const float*)d_in[5];
  const float* br_x   = (const float*)d_in[6];
  const float* We_x   = (const float*)d_in[7];
  const float* att_x  = (const float*)d_in[8];
  const float* bias_x = (const float*)d_in[9];
  const float* Wl_h   = (const float*)d_in[10];
  const float* bl_h   = (const float*)d_in[11];
  const float* Wr_h   = (const float*)d_in[12];
  const float* br_h   = (const float*)d_in[13];
  const float* We_h   = (const float*)d_in[14];
  const float* att_h  = (const float*)d_in[15];
  const float* bias_h = (const float*)d_in[16];
  const float* ln_w   = (const float*)d_in[17];
  const float* ln_b   = (const float*)d_in[18];
  const float* out_w1 = (const float*)d_in[19];
  const float* out_b1 = (const float*)d_in[20];
  const float* out_w2 = (const float*)d_in[21];
  const float* out_b2 = (const float*)d_in[22];

  float* ws = (float*)d_ws;
  long off = 0;
  float* deg     = ws + off; off += NN;
  float* mean_ea = ws + off; off += (long)NN * DEE;
  float* xlx     = ws + off; off += 4L * NN * HID;
  float* xrx     = ws + off; off += 4L * NN * HID;
  float* hlb     = ws + off; off += 4L * NN * HID;
  float* hrb     = ws + off; off += 4L * NN * HID;
  float* logits  = ws + off; off += 8L * EPN * 2;
  float* mbuf    = ws + off; off += 8L * NN * 2;
  float* zbuf    = ws + off; off += 8L * NN * 2;   // zbuf and msg contiguous
  float* msg     = ws + off; off += 8L * NN * HID;
  float* hstate  = ws + off; off += (long)NN * HID; // hstate and cstate contiguous
  float* cstate  = ws + off; off += (long)NN * HID;
  float* tmp     = ws + off; off += (long)NN * HID;

  // self-loop attrs + initial state
  hipMemsetAsync(deg,    0, (NN + (long)NN * DEE) * sizeof(float), stream);
  hipMemsetAsync(hstate, 0, 2L * NN * HID * sizeof(float), stream);
  deg_accum_kernel<<<(EE + 255) / 256, 256, 0, stream>>>(ei, edge_attr, deg, mean_ea);
  mean_div_kernel<<<(NN * DEE + 255) / 256, 256, 0, stream>>>(mean_ea, deg);

  const dim3 ggemm(NN / 16, HID / 32, 4);       // 625 x 2 x 4 (gate batch)
  const dim3 gedge((EPN + 7) / 8, 4, 2);        // 8 edges/block x gate x branch

  for (int t = 0; t < TT; ++t) {
    const float* xt = x_seq + (long)t * NN * FF;   // B == 1
    hipMemsetAsync(zbuf, 0, (8L * NN * 2 + 8L * NN * HID) * sizeof(float), stream);
    hipMemsetAsync(mbuf, 0xFF, 8L * NN * 2 * sizeof(float), stream); // -NaN sentinel

    wmma_gemm64_kernel<FF><<<ggemm, 32, 0, stream>>>(
        xt, Wl_x, bl_x, xlx, (long)FF * HID, HID, (long)NN * HID);
    wmma_gemm64_kernel<FF><<<ggemm, 32, 0, stream>>>(
        xt, Wr_x, br_x, xrx, (long)FF * HID, HID, (long)NN * HID);
    wmma_gemm64_kernel<HID><<<ggemm, 32, 0, stream>>>(
        hstate, Wl_h, bl_h, hlb, (long)HID * HID, HID, (long)NN * HID);
    wmma_gemm64_kernel<HID><<<ggemm, 32, 0, stream>>>(
        hstate, Wr_h, br_h, hrb, (long)HID * HID, HID, (long)NN * HID);

    edge_logits_kernel<<<gedge, 256, 0, stream>>>(ei, edge_attr, mean_ea,
                                                  xlx, xrx, hlb, hrb,
                                                  We_x, We_h, att_x, att_h,
                                                  logits, mbuf);
    edge_accum_kernel<<<gedge, 256, 0, stream>>>(ei, logits, mbuf, xlx, hlb, zbuf, msg);
    msg_finalize_kernel<<<(int)((8L * NN * HID + 255) / 256), 256, 0, stream>>>(
        msg, zbuf, bias_x, bias_h);
    lstm_kernel<<<(NN + 7) / 8, 256, 0, stream>>>(msg, cstate, hstate, ln_w, ln_b);
  }

  // output MLP: gelu(h @ W1 + b1) @ W2 + b2
  wmma_gemm64_kernel<HID><<<dim3(NN / 16, HID / 32, 1), 32, 0, stream>>>(
      hstate, out_w1, out_b1, tmp, 0, 0, 0);
  gelu_kernel<<<(int)(((long)NN * HID + 255) / 256), 256, 0, stream>>>(tmp);
  wmma_gemm_small_kernel<HID, OUTD><<<dim3(NN / 16, 1, 1), 32, 0, stream>>>(
      tmp, out_w2, out_b2, (float*)d_out);
}